// LiftSplatShoot_79121887526974
// MI455X (gfx1250) — compile-verified
//
#include <hip/hip_runtime.h>
#include <hip/hip_bf16.h>

typedef float v4f __attribute__((ext_vector_type(4)));

// LSS constants (match reference)
constexpr int kNX = 200, kNY = 200, kNZ = 1;
constexpr int kC  = 64;
constexpr int kPlane = kNX * kNY;                 // 40000
constexpr int kPtsPerBatch = 6 * 41 * 16 * 44;    // N*D*H*W = 173184
// channel stride in bytes inside the output: NZ*NX*NY floats = 160000 B.
// Fits the signed 24-bit immediate offset of global instructions (±8.3 MB).

// Native CDNA5 no-return f32 global atomic, device scope (resolves in L2),
// with the channel stride folded into the immediate offset field.
#define ATOMIC_ADD_F32_OFF(ptr, val, OFF)                                   \
    asm volatile("global_atomic_add_f32 %0, %1, off offset:" #OFF           \
                 " scope:SCOPE_DEV"                                         \
                 :: "v"(ptr), "v"(val)                                      \
                 : "memory")

__global__ __launch_bounds__(256)
void lss_zero(v4f* __restrict__ out, int n4) {
    int i = blockIdx.x * 256 + threadIdx.x;
    if (i < n4) out[i] = (v4f)0.0f;   // RT stores: keep zeros resident in L2
}

// 16 threads per point; lane l handles channels 4l..4l+3 via one b128 NT load.
__global__ __launch_bounds__(256)
void lss_scatter(const float* __restrict__ x,
                 const int*   __restrict__ geom,
                 float*       __restrict__ out,
                 int npts) {
    const int gtid = blockIdx.x * 256 + threadIdx.x;
    const int p    = gtid >> 4;      // point index
    const int l16  = gtid & 15;      // channel group 0..15
    if (p >= npts) return;

    // Streamed once -> non-temporal; don't evict the output grid from L2.
    const int gx = __builtin_nontemporal_load(geom + 3 * p + 0);
    const int gy = __builtin_nontemporal_load(geom + 3 * p + 1);
    const int gz = __builtin_nontemporal_load(geom + 3 * p + 2);
    // LSS 'kept' filter: dropped points contribute nothing.
    if ((unsigned)gx >= (unsigned)kNX ||
        (unsigned)gy >= (unsigned)kNY ||
        (unsigned)gz >= (unsigned)kNZ) return;

    const int b = p / kPtsPerBatch;

    // Streaming feature read: b128 non-temporal, perfectly coalesced
    // (16 lanes cover the point's contiguous 256 B of channels).
    const v4f f =
        __builtin_nontemporal_load((const v4f*)(x + (size_t)p * kC) + l16);

    const int c0 = l16 * 4;
    // out[((b*C + c)*NZ + gz)*NX*NY + gx*NY + gy]; channel stride = NZ*Plane.
    float* base = out
        + ((size_t)(b * kC + c0) * kNZ + (size_t)gz) * (size_t)kPlane
        + (size_t)gx * kNY + (size_t)gy;

    // 4 channels per thread, 160000-byte stride via immediate offsets.
    ATOMIC_ADD_F32_OFF(base, f.x, 0);
    ATOMIC_ADD_F32_OFF(base, f.y, 160000);
    ATOMIC_ADD_F32_OFF(base, f.z, 320000);
    ATOMIC_ADD_F32_OFF(base, f.w, 480000);
}

extern "C" void kernel_launch(void* const* d_in, const int* in_sizes, int n_in,
                              void* d_out, int out_size, void* d_ws, size_t ws_size,
                              hipStream_t stream) {
    const float* x    = (const float*)d_in[0];
    const int*   geom = (const int*)d_in[1];
    float*       out  = (float*)d_out;

    // Zero the BEV grid every call (atomics accumulate; d_out isn't re-poisoned).
    const int n4 = out_size / 4;                      // 2,560,000 float4
    lss_zero<<<(n4 + 255) / 256, 256, 0, stream>>>((v4f*)out, n4);

    const int npts = in_sizes[1] / 3;                 // 692,736
    const long long threads = (long long)npts * 16;   // 11,083,776
    const int blocks = (int)((threads + 255) / 256);  // 43,296
    lss_scatter<<<blocks, 256, 0, stream>>>(x, geom, out, npts);
}